// GATconv_72464688218548
// MI455X (gfx1250) — compile-verified
//
#include <hip/hip_runtime.h>
#include <hip/hip_bf16.h>
#include <cmath>

// GAT forward for MI455X (gfx1250, wave32, WMMA f32 16x16x4).
// Workspace use: ~134.8 MB of d_ws (Wh, x2, f1/f2, row stats, Wh2).

#define N 4096
#define F_IN 512
#define NHID 512
#define NHEADS 8
#define F_TOT (NHEADS * NHID)   // 4096
#define ALPHA 0.2f
#define NEGINF -9000000000000000.0f

typedef float v2f __attribute__((ext_vector_type(2)));
typedef float v8f __attribute__((ext_vector_type(8)));

// GEMM tiling: block tile 64x256, K-chunk 32, 8 waves of 16x128 each.
// LDS layout is k-pair interleaved so each WMMA fragment is ONE ds_load_b64:
//   lA2[kp][2*m + s] = A[m][2*kp + s]      (s = 0,1)
//   lB2[kp][2*c + s] = B[2*kp + s][c]
#define TM 64
#define TN 256
#define TK 32
#define KP (TK / 2)    // 16 k-pairs per chunk
#define LDA2 164       // 2*TM + 36 : staging stores 8 banks apart, frag halves ~disjoint
#define LDB2 544       // 2*TN + 32 : frag half-waves shifted 32 banks -> disjoint

__device__ __forceinline__ float lrelu(float v) { return v > 0.f ? v : ALPHA * v; }
__device__ __forceinline__ float elu(float v)   { return v > 0.f ? v : expm1f(v); }

// ---------------------------------------------------------------- adj copy
__global__ void k_copy(const float4* __restrict__ src, float4* __restrict__ dst, size_t n4) {
  size_t i = (size_t)blockIdx.x * blockDim.x + threadIdx.x;
  size_t stride = (size_t)gridDim.x * blockDim.x;
  for (; i < n4; i += stride) dst[i] = src[i];
}

// ---------------------------------------------------------------- stage A GEMM: Wh[h] = x @ W_heads[h]
__global__ __launch_bounds__(256) void k_gemm_xw(const float* __restrict__ x,
                                                 const float* __restrict__ Wheads,
                                                 float* __restrict__ Wh) {
  __shared__ float lA2[KP][LDA2];
  __shared__ float lB2[KP][LDB2];
  const int t = threadIdx.x;
  const int lane = t & 31, w = t >> 5;
  const int half = lane >> 4, l16 = lane & 15;
  const int h  = blockIdx.z;
  const int m0 = blockIdx.y * TM;
  const int n0 = blockIdx.x * TN;
  const float* Bp = Wheads + (size_t)h * F_IN * NHID;
  const int wm = (w >> 1) * 16;     // wave row base within tile
  const int wn = (w & 1) * 128;     // wave col base within tile
  const int ar = 2 * (wm + l16);    // A frag column (interleaved units)
  const int br = 2 * (wn + l16);    // B frag column base

  v8f acc[8] = {};

  for (int kc = 0; kc < F_IN; kc += TK) {
    // stage A tile (64x32), k-pair interleaved
    #pragma unroll
    for (int rep = 0; rep < 2; ++rep) {
      int idx = rep * 256 + t;
      int i = idx >> 3, j4 = (idx & 7) * 4;
      float4 v = *(const float4*)(x + (size_t)(m0 + i) * F_IN + kc + j4);
      v2f p0; p0.x = v.x; p0.y = v.y;
      v2f p1; p1.x = v.z; p1.y = v.w;
      *(v2f*)&lA2[(j4 >> 1) + 0][2 * i] = p0;
      *(v2f*)&lA2[(j4 >> 1) + 1][2 * i] = p1;
    }
    // stage B tile (32x256) as 2x2 sub-blocks, k-pair interleaved
    #pragma unroll
    for (int rep = 0; rep < 8; ++rep) {
      int idx = rep * 256 + t;
      int rp = idx >> 7, cp = idx & 127;
      const float* bsrc = Bp + (size_t)(kc + 2 * rp) * NHID + n0 + 2 * cp;
      float2 g0 = *(const float2*)(bsrc);
      float2 g1 = *(const float2*)(bsrc + NHID);
      float4 st; st.x = g0.x; st.y = g1.x; st.z = g0.y; st.w = g1.y;
      *(float4*)&lB2[rp][4 * cp] = st;
    }
    // prefetch next chunk (speculative, one 128B line per thread)
    if (kc + TK < F_IN) {
      __builtin_prefetch(Bp + (size_t)(kc + TK + (t >> 3)) * NHID + n0 + (t & 7) * 32, 0, 0);
      if (t < TM)
        __builtin_prefetch(x + (size_t)(m0 + t) * F_IN + kc + TK, 0, 0);
    }
    __syncthreads();
    #pragma unroll
    for (int ks = 0; ks < 8; ++ks) {
      const int kp = ks * 2 + half;
      v2f a = *(const v2f*)&lA2[kp][ar];
      #pragma unroll
      for (int j = 0; j < 8; ++j) {
        v2f b = *(const v2f*)&lB2[kp][br + j * 32];
        acc[j] = __builtin_amdgcn_wmma_f32_16x16x4_f32(false, a, false, b,
                                                       (short)0, acc[j], false, false);
      }
    }
    __syncthreads();
  }

  float* Cp = Wh + (size_t)h * N * NHID;
  #pragma unroll
  for (int j = 0; j < 8; ++j)
    #pragma unroll
    for (int r = 0; r < 8; ++r) {
      int gm = m0 + wm + r + half * 8;
      int gn = n0 + wn + j * 16 + l16;
      Cp[(size_t)gm * NHID + gn] = acc[j][r];
    }
}

// ---------------------------------------------------------------- f1/f2 per (head,row)
__global__ __launch_bounds__(256) void k_f12(const float* __restrict__ Wh,
                                             const float* __restrict__ aheads,
                                             float* __restrict__ f1, float* __restrict__ f2) {
  int t = threadIdx.x, lane = t & 31, w = t >> 5;
  int idx = blockIdx.x * 8 + w;            // 0..32767
  int h = idx >> 12, n = idx & (N - 1);
  const float* row = Wh + (size_t)h * N * NHID + (size_t)n * NHID;
  const float* a1 = aheads + h * 2 * NHID;
  const float* a2 = a1 + NHID;
  float s1 = 0.f, s2 = 0.f;
  for (int d = lane; d < NHID; d += 32) {
    float v = row[d];
    s1 += v * a1[d];
    s2 += v * a2[d];
  }
  for (int o = 16; o; o >>= 1) {
    s1 += __shfl_down(s1, o, 32);
    s2 += __shfl_down(s2, o, 32);
  }
  if (lane == 0) { f1[h * N + n] = s1; f2[h * N + n] = s2; }
}

// ---------------------------------------------------------------- softmax row stats per (head,row)
__global__ __launch_bounds__(256) void k_rowstats(const float* __restrict__ adj,
                                                  const float* __restrict__ f1,
                                                  const float* __restrict__ f2,
                                                  float* __restrict__ rmax,
                                                  float* __restrict__ rinv) {
  __shared__ float se[N];
  __shared__ float red[256];
  int t = threadIdx.x;
  int n = blockIdx.x, h = blockIdx.y;
  float f1v = f1[h * N + n];
  const float* arow = adj + (size_t)n * N;
  const float* f2h = f2 + h * N;
  float lmax = NEGINF;
  for (int m = t; m < N; m += 256) {
    float e = (arow[m] > 0.f) ? lrelu(f1v + f2h[m]) : NEGINF;
    se[m] = e;
    lmax = fmaxf(lmax, e);
  }
  red[t] = lmax; __syncthreads();
  for (int s = 128; s; s >>= 1) { if (t < s) red[t] = fmaxf(red[t], red[t + s]); __syncthreads(); }
  float mx = red[0]; __syncthreads();
  float ls = 0.f;
  for (int m = t; m < N; m += 256) ls += expf(se[m] - mx);
  red[t] = ls; __syncthreads();
  for (int s = 128; s; s >>= 1) { if (t < s) red[t] += red[t + s]; __syncthreads(); }
  if (t == 0) { rmax[h * N + n] = mx; rinv[h * N + n] = 1.f / red[0]; }
}

// ---------------------------------------------------------------- fused att GEMM: x2[:,h*512..] = elu(att[h] @ Wh[h])
__global__ __launch_bounds__(256) void k_gemm_att(const float* __restrict__ adj,
                                                  const float* __restrict__ Wh,
                                                  const float* __restrict__ f1,
                                                  const float* __restrict__ f2,
                                                  const float* __restrict__ rmax,
                                                  const float* __restrict__ rinv,
                                                  float* __restrict__ x2) {
  __shared__ float lA2[KP][LDA2];
  __shared__ float lB2[KP][LDB2];
  __shared__ float sF1[TM], sRmax[TM], sRinv[TM];
  const int t = threadIdx.x;
  const int lane = t & 31, w = t >> 5;
  const int half = lane >> 4, l16 = lane & 15;
  const int h  = blockIdx.z;
  const int m0 = blockIdx.y * TM;
  const int n0 = blockIdx.x * TN;
  const float* Bp  = Wh + (size_t)h * N * NHID;
  const float* f2h = f2 + h * N;
  const int wm = (w >> 1) * 16;
  const int wn = (w & 1) * 128;
  const int ar = 2 * (wm + l16);
  const int br = 2 * (wn + l16);

  if (t < TM) {
    int gm = m0 + t;
    sF1[t]   = f1[h * N + gm];
    sRmax[t] = rmax[h * N + gm];
    sRinv[t] = rinv[h * N + gm];
  }
  __syncthreads();

  v8f acc[8] = {};

  for (int kc = 0; kc < N; kc += TK) {
    // stage A tile: attention values built on the fly from adj + row stats
    #pragma unroll
    for (int rep = 0; rep < 2; ++rep) {
      int idx = rep * 256 + t;
      int i = idx >> 3, j4 = (idx & 7) * 4;
      int gm = m0 + i, gk = kc + j4;
      float4 av = *(const float4*)(adj + (size_t)gm * N + gk);
      float f1v = sF1[i], rm = sRmax[i], ri = sRinv[i];
      float vq[4] = {av.x, av.y, av.z, av.w};
      float aq[4];
      #pragma unroll
      for (int q = 0; q < 4; ++q) {
        float e = (vq[q] > 0.f) ? lrelu(f1v + f2h[gk + q]) : NEGINF;
        aq[q] = expf(e - rm) * ri;
      }
      v2f p0; p0.x = aq[0]; p0.y = aq[1];
      v2f p1; p1.x = aq[2]; p1.y = aq[3];
      *(v2f*)&lA2[(j4 >> 1) + 0][2 * i] = p0;
      *(v2f*)&lA2[(j4 >> 1) + 1][2 * i] = p1;
    }
    // stage B tile from Wh[h], 2x2 sub-blocks interleaved
    #pragma unroll
    for (int rep = 0; rep < 8; ++rep) {
      int idx = rep * 256 + t;
      int rp = idx >> 7, cp = idx & 127;
      const float* bsrc = Bp + (size_t)(kc + 2 * rp) * NHID + n0 + 2 * cp;
      float2 g0 = *(const float2*)(bsrc);
      float2 g1 = *(const float2*)(bsrc + NHID);
      float4 st; st.x = g0.x; st.y = g1.x; st.z = g0.y; st.w = g1.y;
      *(float4*)&lB2[rp][4 * cp] = st;
    }
    // prefetch next chunk
    if (kc + TK < N) {
      __builtin_prefetch(Bp + (size_t)(kc + TK + (t >> 3)) * NHID + n0 + (t & 7) * 32, 0, 0);
      if (t < TM)
        __builtin_prefetch(adj + (size_t)(m0 + t) * N + kc + TK, 0, 0);
    }
    __syncthreads();
    #pragma unroll
    for (int ks = 0; ks < 8; ++ks) {
      const int kp = ks * 2 + half;
      v2f a = *(const v2f*)&lA2[kp][ar];
      #pragma unroll
      for (int j = 0; j < 8; ++j) {
        v2f b = *(const v2f*)&lB2[kp][br + j * 32];
        acc[j] = __builtin_amdgcn_wmma_f32_16x16x4_f32(false, a, false, b,
                                                       (short)0, acc[j], false, false);
      }
    }
    __syncthreads();
  }

  // epilogue: elu, scatter into x2 at column offset h*NHID
  #pragma unroll
  for (int j = 0; j < 8; ++j)
    #pragma unroll
    for (int r = 0; r < 8; ++r) {
      int gm = m0 + wm + r + half * 8;
      int gn = h * NHID + n0 + wn + j * 16 + l16;
      x2[(size_t)gm * F_TOT + gn] = elu(acc[j][r]);
    }
}

// ---------------------------------------------------------------- Wh2 = x2 @ W_out  (GEMV, NOUT=1)
__global__ __launch_bounds__(256) void k_gemv_wout(const float* __restrict__ x2,
                                                   const float* __restrict__ Wout,
                                                   float* __restrict__ Wh2) {
  __shared__ float red[256];
  int t = threadIdx.x, n = blockIdx.x;
  const float* row = x2 + (size_t)n * F_TOT;
  float s = 0.f;
  for (int j = t; j < F_TOT; j += 256) s += row[j] * Wout[j];
  red[t] = s; __syncthreads();
  for (int st = 128; st; st >>= 1) { if (t < st) red[t] += red[t + st]; __syncthreads(); }
  if (t == 0) Wh2[n] = red[0];
}

// ---------------------------------------------------------------- output attention row + elu + log_softmax(singleton)
__global__ __launch_bounds__(256) void k_final(const float* __restrict__ adj,
                                               const float* __restrict__ Wh2,
                                               const float* __restrict__ aout,
                                               float* __restrict__ outp) {
  __shared__ float se[N];
  __shared__ float red[256];
  __shared__ float red2[256];
  int t = threadIdx.x, n = blockIdx.x;
  float a0 = aout[0], a1 = aout[1];
  float f1v = Wh2[n] * a0;
  const float* arow = adj + (size_t)n * N;
  float lmax = NEGINF;
  for (int m = t; m < N; m += 256) {
    float e = (arow[m] > 0.f) ? lrelu(f1v + Wh2[m] * a1) : NEGINF;
    se[m] = e;
    lmax = fmaxf(lmax, e);
  }
  red[t] = lmax; __syncthreads();
  for (int s = 128; s; s >>= 1) { if (t < s) red[t] = fmaxf(red[t], red[t + s]); __syncthreads(); }
  float mx = red[0]; __syncthreads();
  float lsum = 0.f, lacc = 0.f;
  for (int m = t; m < N; m += 256) {
    float p = expf(se[m] - mx);
    lsum += p;
    lacc += p * Wh2[m];
  }
  red[t] = lsum; red2[t] = lacc; __syncthreads();
  for (int s = 128; s; s >>= 1) {
    if (t < s) { red[t] += red[t + s]; red2[t] += red2[t + s]; }
    __syncthreads();
  }
  if (t == 0) {
    float hp = red2[0] / red[0];
    float r = elu(hp);
    // log_softmax over NOUT==1 axis: r - (r + log(exp(r-r))) == 0 for finite r
    float lse = r + logf(expf(r - r));
    outp[n] = r - lse;
  }
}

// ----------------------------------------------------------------
extern "C" void kernel_launch(void* const* d_in, const int* in_sizes, int n_in,
                              void* d_out, int out_size, void* d_ws, size_t ws_size,
                              hipStream_t stream) {
  (void)in_sizes; (void)n_in; (void)out_size; (void)ws_size;
  const float* adj    = (const float*)d_in[0];
  const float* x      = (const float*)d_in[1];
  const float* Wheads = (const float*)d_in[2];
  const float* aheads = (const float*)d_in[3];
  const float* Wout   = (const float*)d_in[4];
  const float* aout   = (const float*)d_in[5];
  float* out = (float*)d_out;

  float* ws   = (float*)d_ws;
  float* Wh   = ws;                                  // 8*4096*512
  float* x2   = Wh + (size_t)NHEADS * N * NHID;      // 4096*4096
  float* f1   = x2 + (size_t)N * F_TOT;              // 8*4096
  float* f2   = f1 + NHEADS * N;                     // 8*4096
  float* rmax = f2 + NHEADS * N;                     // 8*4096
  float* rinv = rmax + NHEADS * N;                   // 8*4096
  float* Wh2  = rinv + NHEADS * N;                   // 4096

  // output #1: adj passthrough
  k_copy<<<4096, 256, 0, stream>>>((const float4*)adj, (float4*)out, (size_t)N * N / 4);

  // heads pipeline
  k_gemm_xw<<<dim3(NHID / TN, N / TM, NHEADS), 256, 0, stream>>>(x, Wheads, Wh);
  k_f12<<<N, 256, 0, stream>>>(Wh, aheads, f1, f2);
  k_rowstats<<<dim3(N, NHEADS), 256, 0, stream>>>(adj, f1, f2, rmax, rinv);
  k_gemm_att<<<dim3(NHID / TN, N / TM, NHEADS), 256, 0, stream>>>(adj, Wh, f1, f2, rmax, rinv, x2);

  // output layer (NOUT = 1)
  k_gemv_wout<<<N, 256, 0, stream>>>(x2, Wout, Wh2);
  k_final<<<N, 256, 0, stream>>>(adj, Wh2, aout, out + (size_t)N * N);
}